// YOLOv3Layer_27221502722634
// MI455X (gfx1250) — compile-verified
//
#include <hip/hip_runtime.h>

// YOLOv3 decode layer for MI455X (gfx1250, wave32).
// x: (32, 255, 76, 76) f32 NCHW  ->  out: (32, 76*76*3, 85) f32
// Memory-bound: 377 MB traffic -> ~16 us floor @ 23.3 TB/s. Kernel is
// structured so the hot loop is ~5 ops: coalesced b32 load (full 128B line
// per wave), v_tanh_f32 sigmoid, linear ds_store (stride 255 % 64 banks ->
// conflict-free). Special fields (12/255 channels) patched in a tiny
// uniform-branch fix-up pass. Output drained with async LDS->global b128.

#define YB 32
#define YW 76
#define YHW 5776       // 76*76
#define YNB 3
#define YSKIP 85
#define YNCH 255       // 3*85
#define CHUNK 64       // hw positions per block
#define NCHUNK 91      // ceil(5776/64); last chunk has 16

__device__ __forceinline__ float fast_sigmoid(float x) {
  // sigmoid(x) = 0.5*tanh(x/2) + 0.5 : single gfx1250 TRANS op + fma
#if __has_builtin(__builtin_amdgcn_tanhf)
  return __builtin_amdgcn_tanhf(0.5f * x) * 0.5f + 0.5f;
#else
  float t;
  asm("v_tanh_f32 %0, %1" : "=v"(t) : "v"(0.5f * x));
  return t * 0.5f + 0.5f;
#endif
}

__device__ __forceinline__ float fast_exp(float x) {
  return __builtin_amdgcn_exp2f(x * 1.44269504088896340736f);
}

__global__ __launch_bounds__(256) void yolo_decode_kernel(
    const float* __restrict__ x, float* __restrict__ out) {
  // Tile staged in exact output layout: [hw_local][c] with c = box*85+f.
  __shared__ float lds[CHUNK * YNCH];  // 16320 floats = 65280 B

  const int b     = blockIdx.x / NCHUNK;
  const int chunk = blockIdx.x % NCHUNK;
  const int hw0   = chunk * CHUNK;
  const int rem   = YHW - hw0;
  const int hwN   = (rem < CHUNK) ? rem : CHUNK;

  const int tid  = threadIdx.x;
  const int lane = tid & (CHUNK - 1);  // hw offset within tile
  const int grp  = tid >> 6;           // 0..3

  const float* xin = x + (size_t)b * YNCH * YHW + hw0;

  // ---- Phase 1: sigmoid everything; purely linear addressing ----
  if (lane < hwN) {
    const float* xp = xin + lane;
    float* lp = &lds[lane * YNCH];
#pragma unroll 8
    for (int c = grp; c < YNCH; c += 4) {
      lp[c] = fast_sigmoid(xp[c * YHW]);
    }
  }
  __syncthreads();

  // ---- Phase 2: patch the 12 special channels (f<4 per box) ----
  // spec = box*4+f; spec % 4 == grp, so f (and all branches) are
  // wave-uniform and loop-invariant.
  if (lane < hwN) {
    const int hw     = hw0 + lane;
    const float colf = (float)(hw % YW);
    const float rowf = (float)(hw / YW);
    const float* xp  = xin + lane;
    const int f      = grp;  // spec & 3
    for (int spec = grp; spec < 12; spec += 4) {
      const int box = spec >> 2;          // 0,1,2 over the iterations
      const int c   = box * YSKIP + f;
      const int idx = lane * YNCH + c;
      float r;
      if (f == 0) {
        r = (lds[idx] + colf) * (1.0f / (float)YW);
      } else if (f == 1) {
        r = (lds[idx] + rowf) * (1.0f / (float)YW);
      } else {
        const float e  = fast_exp(xp[c * YHW]);
        const float sx = (box == 0) ? 116.0f / 608.0f
                       : (box == 1) ? 156.0f / 608.0f : 373.0f / 608.0f;
        const float sy = (box == 0) ?  90.0f / 608.0f
                       : (box == 1) ? 198.0f / 608.0f : 326.0f / 608.0f;
        r = ((f == 2) ? sx : sy) * e;
      }
      lds[idx] = r;
    }
  }
  __syncthreads();

  // ---- Phase 3: tile is output-contiguous -> drain straight out ----
  float* dst     = out + ((size_t)b * YHW + hw0) * YNCH;
  const int nvec = (hwN * YNCH) >> 2;  // float4 count (4080 or 1020)

#if __has_builtin(__builtin_amdgcn_global_store_async_from_lds_b128)
  typedef __attribute__((ext_vector_type(4))) int v4i;
  typedef __attribute__((address_space(1))) v4i* gp_t;  // global int4*
  typedef __attribute__((address_space(3))) v4i* lp_t;  // LDS int4*
  for (int i = tid; i < nvec; i += 256) {
    __builtin_amdgcn_global_store_async_from_lds_b128(
        (gp_t)(void*)(dst + 4 * (size_t)i), (lp_t)(void*)&lds[4 * i], 0, 0);
  }
#if __has_builtin(__builtin_amdgcn_s_wait_asynccnt)
  __builtin_amdgcn_s_wait_asynccnt(0);
#else
  asm volatile("s_wait_asynccnt 0" ::: "memory");
#endif
#else
  for (int i = tid; i < nvec; i += 256) {
    ((float4*)dst)[i] = ((const float4*)lds)[i];
  }
#endif
}

extern "C" void kernel_launch(void* const* d_in, const int* in_sizes, int n_in,
                              void* d_out, int out_size, void* d_ws, size_t ws_size,
                              hipStream_t stream) {
  (void)in_sizes; (void)n_in; (void)out_size; (void)d_ws; (void)ws_size;
  const float* x = (const float*)d_in[0];
  float* out     = (float*)d_out;
  dim3 grid(YB * NCHUNK);  // 32 images * 91 hw-chunks = 2912 blocks
  yolo_decode_kernel<<<grid, 256, 0, stream>>>(x, out);
}